// GumbelVectorQuantizer_Predictive_Parallel_83124797047449
// MI455X (gfx1250) — compile-verified
//
#include <hip/hip_runtime.h>
#include <hip/hip_bf16.h>

typedef __attribute__((ext_vector_type(2))) float v2f;
typedef __attribute__((ext_vector_type(8))) float v8f;

#define G_     2
#define M_     1024
#define IPG    128
#define KCTX_  7
#define BB     8
#define TT     2048
#define CC     256
#define NTOT   (BB * TT)
#define ALPHA_ 10.0f

// B matrices are stored k-pair packed: [kpair][col*2 + klow], global pair-row
// stride 256 floats (contiguous), LDS pair-row stride 288 floats (pad so the
// two 16-lane halves of a wave hit disjoint bank sets).
#define LDB_P  288

// ---------------- workspace layout (floats) ----------------
#define WS_ZBUF 0
#define WS_BC   (WS_ZBUF + NTOT * G_ * IPG)          // G * 448 kpair * 256
#define WS_B1   (WS_BC + G_ * 448 * 256)             // G * 128 kpair * 256
#define WS_B2   (WS_B1 + G_ * 128 * 256)
#define WS_ET   (WS_B2 + G_ * 128 * 256)             // G * 8 mc * 64 dpair * 256
#define WS_EE   (WS_ET + G_ * 8 * 64 * 256)

// ---------------- async global -> LDS staging ----------------
#if __has_builtin(__builtin_amdgcn_global_load_async_to_lds_b128) && \
    __has_builtin(__builtin_amdgcn_s_wait_asynccnt)
typedef int v4i_ __attribute__((vector_size(16)));
typedef __attribute__((address_space(1))) v4i_ as1_v4i;
typedef __attribute__((address_space(3))) v4i_ as3_v4i;
__device__ __forceinline__ void copy_b128(const float* gsrc, float* ldst) {
  __builtin_amdgcn_global_load_async_to_lds_b128(
      (as1_v4i*)(uintptr_t)gsrc, (as3_v4i*)(unsigned)(uintptr_t)ldst, 0, 0);
}
#define ASYNC_WAIT(n) __builtin_amdgcn_s_wait_asynccnt(n)
#else
__device__ __forceinline__ void copy_b128(const float* gsrc, float* ldst) {
  *(float4*)ldst = *(const float4*)gsrc;
}
#define ASYNC_WAIT(n) ((void)0)
#endif

// stage nkp packed k-pair rows (256 floats each) into LDS (stride LDB_P)
__device__ __forceinline__ void stage_chunk(const float* __restrict__ gsrc,
                                            float* ldst, int nkp, int tid) {
  const int units = nkp * 64;               // 16B units
  for (int u = tid; u < units; u += 256) {
    int row = u >> 6, seg = u & 63;
    copy_b128(gsrc + row * 256 + seg * 4, ldst + row * LDB_P + seg * 4);
  }
}

// -------- 128x128 macro-tile GEMM core: 8 waves, each wave 32x64 ----------
__device__ __forceinline__ void gemm_chunk(const float* A, int lda,
                                           const float* Bp, int Kc,
                                           int lane, int wm, int wn,
                                           v8f acc[2][4]) {
  const int koff = (lane >> 4) * 2;         // lanes 16..31 hold K+2,K+3
  const int l15  = lane & 15;
  for (int kk = 0; kk < Kc; kk += 4) {
    v2f a[2];
#pragma unroll
    for (int rs = 0; rs < 2; ++rs)
      a[rs] = *(const v2f*)&A[(wm * 32 + rs * 16 + l15) * lda + kk + koff];
    v2f b[4];
#pragma unroll
    for (int cs = 0; cs < 4; ++cs)
      b[cs] = *(const v2f*)&Bp[((kk + koff) >> 1) * LDB_P +
                               (wn * 64 + cs * 16 + l15) * 2];
#pragma unroll
    for (int rs = 0; rs < 2; ++rs)
#pragma unroll
      for (int cs = 0; cs < 4; ++cs)
        acc[rs][cs] = __builtin_amdgcn_wmma_f32_16x16x4_f32(
            false, a[rs], false, b[cs], (short)0, acc[rs][cs], false, false);
  }
}

__device__ __forceinline__ void zero_acc(v8f acc[2][4]) {
#pragma unroll
  for (int rs = 0; rs < 2; ++rs)
#pragma unroll
    for (int cs = 0; cs < 4; ++cs)
      acc[rs][cs] = (v8f){0.f, 0.f, 0.f, 0.f, 0.f, 0.f, 0.f, 0.f};
}

// ---------------- kernel 0: pack weights + ||e||^2 ----------------
__global__ void gvq_prep(const float* __restrict__ ctx_w,
                         const float* __restrict__ fuse1_w,
                         const float* __restrict__ fuse2_w,
                         const float* __restrict__ emb,
                         float* __restrict__ ws) {
  float* Bc = ws + WS_BC;
  float* B1 = ws + WS_B1;
  float* B2 = ws + WS_B2;
  float* Et = ws + WS_ET;
  float* ee = ws + WS_EE;
  int i = blockIdx.x * 256 + threadIdx.x;
  const int R0 = G_ * 448 * 256;            // 229,376
  const int R1 = G_ * 128 * 256;            //  65,536
  const int R2 = G_ * 128 * 256;
  const int R3 = G_ * 8 * 64 * 256;         // 262,144
  const int R4 = G_ * 1024;
  if (i < R0) {
    int g = i / (448 * 256), r = i % (448 * 256);
    int kpair = r >> 8, rr = r & 255;
    int col = rr >> 1, klow = rr & 1;
    int kk = kpair * 2 + klow;              // 0..895
    int tap = kk >> 7, ci = kk & 127;
    Bc[i] = ctx_w[((g * 128 + col) * 128 + ci) * KCTX_ + tap];
  } else if ((i -= R0) < R1) {
    int g = i / (128 * 256), r = i % (128 * 256);
    int kpair = r >> 8, rr = r & 255;
    int col = rr >> 1, kk = kpair * 2 + (rr & 1);
    B1[i] = fuse1_w[(g * 128 + col) * 256 + kk];
  } else if ((i -= R1) < R2) {
    int g = i / (128 * 256), r = i % (128 * 256);
    int kpair = r >> 8, rr = r & 255;
    int col = rr >> 1, kk = kpair * 2 + (rr & 1);
    B2[i] = fuse2_w[(g * 128 + col) * 256 + kk];
  } else if ((i -= R2) < R3) {
    int g = i / (8 * 64 * 256), r = i % (8 * 64 * 256);
    int mc = r / 16384, r2 = r % 16384;
    int dpair = r2 >> 8, rr = r2 & 255;
    int mloc = rr >> 1, d = dpair * 2 + (rr & 1);
    int m = mc * 128 + mloc;
    Et[i] = emb[(g * 1024 + m) * 128 + d];
  } else if ((i -= R3) < R4) {
    int g = i >> 10, m = i & 1023;
    const float* e = &emb[(g * 1024 + m) * 128];
    float s = 0.f;
    for (int d = 0; d < 128; ++d) s += e[d] * e[d];
    ee[i] = s;
  }
}

// ------- kernel 1: context conv + fuse1 + PReLU + fuse2 + PReLU -> z -------
// grid (T/128, B, G), block 256 (8 waves)
__global__ void gvq_ctx_fuse(const float* __restrict__ x,
                             const float* __restrict__ ar,
                             const float* __restrict__ ws_in,
                             const float* __restrict__ prelu1,
                             const float* __restrict__ prelu2,
                             float* __restrict__ ws_out) {
  extern __shared__ float smem[];
  float* sAr = smem;                  // 136 * 132 = 17,952
  float* sX  = sAr + 136 * 132;       // 128 * 132 = 16,896
  float* sC  = sX + 128 * 132;        // 16,896 (ctx, then m)
  float* sB0 = sC + 128 * 132;        // two buffers of 32 * 288 = 9,216 each
  const int BUF1 = 32 * LDB_P;

  const float* Bc = ws_in + WS_BC;
  const float* B1 = ws_in + WS_B1;
  const float* B2 = ws_in + WS_B2;
  float* zbuf = ws_out + WS_ZBUF;

  const int tid  = threadIdx.x;
  const int lane = tid & 31;
  const int wave = tid >> 5;
  const int wm = wave >> 1, wn = wave & 1;
  const int t0 = blockIdx.x * 128;
  const int b  = blockIdx.y;
  const int g  = blockIdx.z;
  const int l15 = lane & 15;
  const int rhi = (lane & 16) ? 8 : 0;

  // ---- stage ar tile (rows t0-7 .. t0+126 of ar[:, :-1, :]) ----
  for (int u = tid; u < 134 * 32; u += 256) {
    int rr = u >> 5, seg = u & 31;
    int t = t0 - KCTX_ + rr;
    float* dst = &sAr[rr * 132 + seg * 4];
    if (t >= 0 && t < TT - 1)
      copy_b128(&ar[((size_t)(b * TT + t)) * CC + g * IPG + seg * 4], dst);
    else
      *(float4*)dst = make_float4(0.f, 0.f, 0.f, 0.f);
  }
  // ---- stage x tile ----
  for (int u = tid; u < 128 * 32; u += 256) {
    int rr = u >> 5, seg = u & 31;
    copy_b128(&x[((size_t)(b * TT + t0 + rr)) * CC + g * IPG + seg * 4],
              &sX[rr * 132 + seg * 4]);
  }

  v8f acc[2][4];
  zero_acc(acc);

  // ---- GEMM1: ctx = im2col(ar) @ Wc^T, 14 chunks of K=64, double-buffered --
  const float* BcG = Bc + (size_t)g * 448 * 256;
  stage_chunk(BcG, sB0, 32, tid);
  for (int c = 0; c < 14; ++c) {
    if (c < 13) { stage_chunk(BcG + (c + 1) * 32 * 256,
                              sB0 + ((c + 1) & 1) * BUF1, 32, tid);
                  ASYNC_WAIT(8); } else { ASYNC_WAIT(0); }
    __syncthreads();
    gemm_chunk(sAr + (c >> 1) * 132 + (c & 1) * 64, 132,
               sB0 + (c & 1) * BUF1, 64, lane, wm, wn, acc);
    __syncthreads();
  }
  const float* B1G = B1 + (size_t)g * 128 * 256;
  stage_chunk(B1G, sB0, 32, tid);           // prefetch fuse1 chunk0
#pragma unroll
  for (int rs = 0; rs < 2; ++rs)
#pragma unroll
    for (int cs = 0; cs < 4; ++cs)
#pragma unroll
      for (int v = 0; v < 8; ++v)
        sC[(wm * 32 + rs * 16 + v + rhi) * 132 + wn * 64 + cs * 16 + l15] =
            acc[rs][cs][v];
  __syncthreads();

  // ---- GEMM2: m = W1 @ [x; ctx], 4 chunks of K=64, PReLU ----
  const float a1 = prelu1[g];
  zero_acc(acc);
  for (int c = 0; c < 4; ++c) {
    if (c < 3) { stage_chunk(B1G + (c + 1) * 32 * 256,
                             sB0 + ((c + 1) & 1) * BUF1, 32, tid);
                 ASYNC_WAIT(8); } else { ASYNC_WAIT(0); }
    __syncthreads();
    const float* Ab = ((c < 2) ? sX : sC) + (c & 1) * 64;
    gemm_chunk(Ab, 132, sB0 + (c & 1) * BUF1, 64, lane, wm, wn, acc);
    __syncthreads();
  }
  const float* B2G = B2 + (size_t)g * 128 * 256;
  stage_chunk(B2G, sB0, 32, tid);           // prefetch fuse2 chunk0
#pragma unroll
  for (int rs = 0; rs < 2; ++rs)
#pragma unroll
    for (int cs = 0; cs < 4; ++cs)
#pragma unroll
      for (int v = 0; v < 8; ++v) {
        float m = acc[rs][cs][v];
        m = (m >= 0.f) ? m : a1 * m;
        sC[(wm * 32 + rs * 16 + v + rhi) * 132 + wn * 64 + cs * 16 + l15] = m;
      }
  __syncthreads();

  // ---- GEMM3: z = W2 @ [x; m], PReLU, store ----
  const float a2 = prelu2[g];
  zero_acc(acc);
  for (int c = 0; c < 4; ++c) {
    if (c < 3) { stage_chunk(B2G + (c + 1) * 32 * 256,
                             sB0 + ((c + 1) & 1) * BUF1, 32, tid);
                 ASYNC_WAIT(8); } else { ASYNC_WAIT(0); }
    __syncthreads();
    const float* Ab = ((c < 2) ? sX : sC) + (c & 1) * 64;
    gemm_chunk(Ab, 132, sB0 + (c & 1) * BUF1, 64, lane, wm, wn, acc);
    __syncthreads();
  }
#pragma unroll
  for (int rs = 0; rs < 2; ++rs)
#pragma unroll
    for (int cs = 0; cs < 4; ++cs)
#pragma unroll
      for (int v = 0; v < 8; ++v) {
        float zv = acc[rs][cs][v];
        zv = (zv >= 0.f) ? zv : a2 * zv;
        int row = wm * 32 + rs * 16 + v + rhi;
        int col = wn * 64 + cs * 16 + l15;
        size_t n = (size_t)b * TT + t0 + row;
        zbuf[(n * G_ + g) * IPG + col] = zv;
      }
}

// ------- kernel 2: scores (WMMA) + gumbel + argmax + embedding gather -------
// grid (N/128, G), block 256
__global__ void gvq_score_gather(const float* __restrict__ ws_in,
                                 const float* __restrict__ gumbel,
                                 const float* __restrict__ emb,
                                 float* __restrict__ out) {
  extern __shared__ float smem[];
  float* sZ = smem;                   // 128 * 132 = 16,896
  float* sB0 = sZ + 128 * 132;        // two buffers of 64 * 288 = 18,432 each
  const int BUF2 = 64 * LDB_P;
  int* sIdx = (int*)(sB0 + 2 * BUF2);

  const float* zbuf = ws_in + WS_ZBUF;
  const float* Et   = ws_in + WS_ET;
  const float* ee   = ws_in + WS_EE;

  const int tid  = threadIdx.x;
  const int lane = tid & 31;
  const int wave = tid >> 5;
  const int wm = wave >> 1, wn = wave & 1;
  const int n0 = blockIdx.x * 128;
  const int g  = blockIdx.y;
  const int l15 = lane & 15;
  const int rhi = (lane & 16) ? 8 : 0;

  // stage z tile + codebook chunk 0
  for (int u = tid; u < 128 * 32; u += 256) {
    int rr = u >> 5, seg = u & 31;
    copy_b128(&zbuf[((size_t)(n0 + rr) * G_ + g) * IPG + seg * 4],
              &sZ[rr * 132 + seg * 4]);
  }
  const float* EtG = Et + (size_t)g * 8 * 64 * 256;
  stage_chunk(EtG, sB0, 64, tid);

  float best[16];
  int bidx[16];
#pragma unroll
  for (int s = 0; s < 16; ++s) { best[s] = -INFINITY; bidx[s] = 0; }

  for (int mc = 0; mc < 8; ++mc) {
    if (mc < 7) { stage_chunk(EtG + (mc + 1) * 16384,
                              sB0 + ((mc + 1) & 1) * BUF2, 64, tid);
                  ASYNC_WAIT(16); } else { ASYNC_WAIT(0); }
    // prefetch this chunk's gumbel slab while the GEMM runs
    {
      int rr = tid >> 1, part = tid & 1;
      __builtin_prefetch(&gumbel[(size_t)(n0 + rr) * (G_ * M_) + g * M_ +
                                 mc * 128 + part * 64], 0, 0);
    }
    __syncthreads();

    v8f acc[2][4];
    zero_acc(acc);
    gemm_chunk(sZ, 132, sB0 + (mc & 1) * BUF2, 128, lane, wm, wn, acc);

#pragma unroll
    for (int rs = 0; rs < 2; ++rs)
#pragma unroll
      for (int cs = 0; cs < 4; ++cs)
#pragma unroll
        for (int v = 0; v < 8; ++v) {
          int row = wm * 32 + rs * 16 + v + rhi;
          int m = mc * 128 + wn * 64 + cs * 16 + l15;
          size_t n = (size_t)(n0 + row);
          float sc = 2.0f * ALPHA_ * acc[rs][cs][v]
                   - ALPHA_ * ee[g * M_ + m]
                   + gumbel[n * (G_ * M_) + g * M_ + m];
          int s = rs * 8 + v;
          if (sc > best[s]) { best[s] = sc; bidx[s] = m; }
        }
    __syncthreads();
  }

  // cross-lane argmax within each 16-lane half (same row), first-index ties
#pragma unroll
  for (int s = 0; s < 16; ++s) {
    float v = best[s];
    int id = bidx[s];
    for (int off = 1; off < 16; off <<= 1) {
      float ov = __shfl_xor(v, off, 32);
      int oi = __shfl_xor(id, off, 32);
      if (ov > v || (ov == v && oi < id)) { v = ov; id = oi; }
    }
    if (l15 == 0) {
      int rs = s >> 3, vv = s & 7;
      sIdx[wm * 32 + rs * 16 + vv + rhi] = id;
    }
  }
  __syncthreads();

  // gather embedding rows -> out (B,T,G*D)
  for (int i = tid; i < 128 * 128; i += 256) {
    int rr = i >> 7, d = i & 127;
    int idx = sIdx[rr];
    out[(size_t)(n0 + rr) * (G_ * IPG) + g * IPG + d] =
        emb[((size_t)g * M_ + idx) * IPG + d];
  }
}

extern "C" void kernel_launch(void* const* d_in, const int* in_sizes, int n_in,
                              void* d_out, int out_size, void* d_ws, size_t ws_size,
                              hipStream_t stream) {
  const float* x        = (const float*)d_in[0];
  const float* ar       = (const float*)d_in[1];
  const float* gumbel   = (const float*)d_in[2];
  const float* ctx_w    = (const float*)d_in[3];
  const float* fuse1_w  = (const float*)d_in[4];
  const float* prelu1_a = (const float*)d_in[5];
  const float* fuse2_w  = (const float*)d_in[6];
  const float* prelu2_a = (const float*)d_in[7];
  const float* emb      = (const float*)d_in[8];
  float* out = (float*)d_out;
  float* ws  = (float*)d_ws;

  gvq_prep<<<2440, 256, 0, stream>>>(ctx_w, fuse1_w, fuse2_w, emb, ws);

  size_t smem1 =
      (size_t)(136 * 132 + 128 * 132 + 128 * 132 + 2 * 32 * LDB_P) * 4;
  dim3 grid1(TT / 128, BB, G_);
  gvq_ctx_fuse<<<grid1, 256, smem1, stream>>>(x, ar, ws, prelu1_a, prelu2_a, ws);

  size_t smem2 = (size_t)(128 * 132 + 2 * 64 * LDB_P) * 4 + 128 * 4;
  dim3 grid2(NTOT / 128, G_);
  gvq_score_gather<<<grid2, 256, smem2, stream>>>(ws, gumbel, emb, out);
}